// projection_layer_n_20091857011277
// MI455X (gfx1250) — compile-verified
//
#include <hip/hip_runtime.h>
#include <stdint.h>
#include <stddef.h>

// Problem constants (from reference)
#define B_      2
#define N_IN_   262144
#define D_      32
#define N_OUT_  65536
#define K_      4
#define NK_     9
#define P_      (B_ * N_OUT_ * K_)      // 524288 output points
#define WAVES_  8                       // waves per block (256 threads, wave32)
#define LOG2E_  1.4426950408889634f

// ---- CDNA5 feature guards -------------------------------------------------
#ifndef USE_ASYNC_GATHER
#define USE_ASYNC_GATHER 1
#endif

#if USE_ASYNC_GATHER && __has_builtin(__builtin_amdgcn_global_load_async_to_lds_b128) && __has_builtin(__builtin_amdgcn_s_wait_asynccnt)
#define HAVE_ASYNC 1
#else
#define HAVE_ASYNC 0
#endif

// Raw v_exp_f32 on the device pass; clang portable builtin on the host pass
// (host pass still parses the kernel body — must not reference host-only libm).
#if defined(__HIP_DEVICE_COMPILE__) && __has_builtin(__builtin_amdgcn_exp2f)
#define EXP2F(v) __builtin_amdgcn_exp2f(v)
#else
#define EXP2F(v) __builtin_exp2f(v)
#endif

// Exact pointee type from the builtin's signature:
//   'int __attribute__((vector_size(4*sizeof(int)))) __device__ *'
typedef int v4i_ __attribute__((vector_size(4 * sizeof(int))));
typedef __attribute__((address_space(1))) v4i_ g_v4i;   // global (__device__)
typedef __attribute__((address_space(3))) v4i_ l_v4i;   // LDS (__shared__)

// One wave per output point; lane == feature channel (D == warpSize == 32).
__global__ __launch_bounds__(256) void projection_gather_kernel(
    const float* __restrict__ x,      // (B, N_IN, D)
    const float* __restrict__ ci,     // (2, B, N_IN)
    const float* __restrict__ co,     // (2, B, N_OUT, K) -> flat (2, P)
    const float* __restrict__ sigma,  // (D)
    const int*   __restrict__ nbr,    // (B, N_OUT, K, NK) -> flat (P, NK)
    float* __restrict__ out)          // (B, N_OUT*K, D) -> flat (P, D)
{
#if HAVE_ASYNC
    __shared__ float smem[WAVES_ * NK_ * D_];   // 9216 B of 320 KB/WGP
#endif
    const int lane = (int)(threadIdx.x & 31u);
    // Force wave-uniform values into SGPRs so address math + coord/index loads
    // go down the scalar (SMEM) path, keeping VMEM free for feature gathers.
    const int wave = __builtin_amdgcn_readfirstlane((int)(threadIdx.x >> 5));
    const int p    = __builtin_amdgcn_readfirstlane((int)blockIdx.x) * WAVES_ + wave;

    // Per-lane (per-channel) weight exponent scale: w = exp2(d2 * a[lane])
    const float s = sigma[lane];
    const float a = -LOG2E_ / (2.0f * s * s);

    const int   b   = p >> 18;          // p / (N_OUT*K), N_OUT*K = 2^18
    const float cox = co[p];
    const float coy = co[P_ + p];

    // 9 neighbor indices (wave-uniform addresses)
    int idx[NK_];
#pragma unroll
    for (int j = 0; j < NK_; ++j) idx[j] = nbr[p * NK_ + j];

    // Wave-uniform squared distances
    float d2[NK_];
#pragma unroll
    for (int j = 0; j < NK_; ++j) {
        const float cix = ci[b * N_IN_ + idx[j]];
        const float ciy = ci[B_ * N_IN_ + b * N_IN_ + idx[j]];
        const float dx = cox - cix;
        const float dy = coy - ciy;
        d2[j] = dx * dx + dy * dy;
    }

    float num = 0.0f, den = 0.0f;

#if HAVE_ASYNC
    // --- CDNA5 async gather: stage 9 rows x 128B into LDS (ASYNCcnt path) ---
    {
        const int r0 = lane >> 3;                 // row within group of 4
        const int c  = lane & 7;                  // 16B chunk within row
        const char* gbase = (const char*)x;
        char*       lbase = (char*)smem + (size_t)wave * (NK_ * D_ * 4);
#pragma unroll
        for (int t = 0; t < 3; ++t) {
            const int r = t * 4 + r0;             // rows 0..11, mask >= 9
            if (r < NK_) {
                const char* gp = gbase +
                    ((size_t)(b * N_IN_ + idx[r]) * (D_ * 4)) + (size_t)(c * 16);
                char* lp = lbase + r * (D_ * 4) + c * 16;
                __builtin_amdgcn_global_load_async_to_lds_b128(
                    (g_v4i*)gp, (l_v4i*)lp, 0, 0);
            }
        }
        __builtin_amdgcn_s_wait_asynccnt(0);      // s_wait_asynccnt 0
        const float* my = smem + wave * (NK_ * D_);
#pragma unroll
        for (int j = 0; j < NK_; ++j) {
            const float w = EXP2F(d2[j] * a);
            num = fmaf(w, my[j * D_ + lane], num);
            den += w;
        }
    }
#else
    // --- Direct path: 9 coalesced 128B row gathers straight to VGPRs ---
    {
        float f[NK_];
#pragma unroll
        for (int j = 0; j < NK_; ++j)
            f[j] = x[(size_t)(b * N_IN_ + idx[j]) * D_ + lane];
#pragma unroll
        for (int j = 0; j < NK_; ++j) {
            const float w = EXP2F(d2[j] * a);
            num = fmaf(w, f[j], num);
            den += w;
        }
    }
#endif

    out[(size_t)p * D_ + lane] = num / (den + 1e-9f);
}

extern "C" void kernel_launch(void* const* d_in, const int* in_sizes, int n_in,
                              void* d_out, int out_size, void* d_ws, size_t ws_size,
                              hipStream_t stream) {
    const float* x  = (const float*)d_in[0];   // x_level_in
    const float* ci = (const float*)d_in[1];   // coords_in
    const float* co = (const float*)d_in[2];   // coords_out
    const float* sg = (const float*)d_in[3];   // sigma
    const int*   nb = (const int*)d_in[4];     // neighbor_idx
    float* out = (float*)d_out;

    dim3 grid(P_ / WAVES_);                    // 65536 blocks, exact (no tail)
    dim3 block(256);                           // 8 waves of 32
    projection_gather_kernel<<<grid, block, 0, stream>>>(x, ci, co, sg, nb, out);
}